// QuantizedLinear_42176578847200
// MI455X (gfx1250) — compile-verified
//
#include <hip/hip_runtime.h>
#include <hip/hip_bf16.h>
#include <stdint.h>

// ---------------------------------------------------------------------------
// Trellis-walk LUT dequantization (QTIP-style) for MI455X / gfx1250.
//
// W[r,c] = 0.011 * sign_l[r] * lut[walks[tile, (i*16+j)/2], (i*16+j)%2] * sign_r[c]
//   M = N = 8192, 16x16 tiles over a 512x512 tile grid, 128 steps * 2 vals.
//
// Bottleneck analysis (MI455X):
//   streaming: 128 MiB walks read + 256 MiB NT stores  -> ~17 us @ 23.3 TB/s
//   gathers:   33.5M random 8B loads from a 512 KB LUT -> 64-128B sector per
//              gather over L2->WGP fabric (~2-4 GB) = the real limiter.
// CDNA5 fix: 320 KB LDS per WGP. Keep HALF the LUT (256 KB) resident in LDS,
// staged once per block with async global->LDS; route each gather branchlessly
// through a per-lane FLAT load (LDS or global chosen by index). Halves the
// dominant gather sector traffic. Walks are double-buffered via async-to-LDS.
// ---------------------------------------------------------------------------

typedef float v2f __attribute__((ext_vector_type(2)));

#define NUM_STRIPS   8192       // (512 tile rows) * (512/32 strips per row)
#define STRIP_INTS   4096       // 32 tiles * 128 steps
#define LUT_LDS_N    32768      // LUT entries resident in LDS (x8B = 256 KB)
#define LUT_LDS_B    262144
#define WL_B         16384      // one walks strip = 16 KB
#define SMEM_BYTES   (LUT_LDS_B + 2 * WL_B)   // 294912 = 288 KB of 320 KB LDS
#define W_SCALE_F    0.011f

__device__ __forceinline__ uint32_t lds_off_u32(const void* p) {
    // generic->LDS addrspacecast on AMDGPU is a low-32-bit truncation: the low
    // 32 bits of a generic pointer to LDS are the wave-relative byte offset,
    // which is exactly what VDST of global_load_async_to_lds expects.
    return (uint32_t)(uint64_t)(size_t)p;
}

__device__ __forceinline__ void async_b128(uint32_t lds_off, const void* gaddr) {
    // per-lane 16B global -> LDS copy, tracked by ASYNCcnt
    asm volatile("global_load_async_to_lds_b128 %0, %1, off"
                 :: "v"(lds_off), "v"(gaddr) : "memory");
}

__device__ __forceinline__ void async_stage_strip(const int* __restrict__ gsrc,
                                                  uint32_t lds_base, int tid) {
    const char* g = (const char*)gsrc;
#pragma unroll
    for (int p = 0; p < 2; ++p)     // 512 thr * 2 * 16B = 16 KB
        async_b128(lds_base + (uint32_t)(p * 8192 + tid * 16),
                   g + p * 8192 + tid * 16);
}

extern "C" __global__ __launch_bounds__(512)
void trellis_dequant_kernel(const int*  __restrict__ walks,
                            const v2f*  __restrict__ lut2,     // 65536 x (2 floats)
                            const float* __restrict__ sign_l,  // 8192
                            const v2f*  __restrict__ sign_r2,  // 4096 v2f
                            v2f*        __restrict__ out2) {   // 8192 x 4096 v2f
    extern __shared__ char smem[];           // [0,256K) LUT | 2 x 16K walks

    const int tid    = threadIdx.x;          // 0..511 (16 wave32)
    const int s0     = blockIdx.x;
    const int stride = gridDim.x;

    // scalar LDS bookkeeping only (no aggregates of LDS-derived pointers:
    // those get promoted to static initializers with addrspacecast -> lld err)
    const uint32_t smem_lds = lds_off_u32(smem);
    const uint32_t wl_lds0  = smem_lds + LUT_LDS_B;     // walks buffer 0
    const uint32_t wl_lds1  = wl_lds0 + WL_B;           // walks buffer 1
    const int*     wl_base  = (const int*)(smem + LUT_LDS_B);

    // --- one-time: stage lower half of LUT (256 KB) into LDS, async ---------
#pragma unroll
    for (int c = 0; c < 32; ++c)             // 32 * 512 thr * 16B = 256 KB
        async_b128(smem_lds + (uint32_t)(c * 8192 + tid * 16),
                   (const char*)lut2 + c * 8192 + tid * 16);

    int cur = 0;
    if (s0 < NUM_STRIPS)
        async_stage_strip(walks + (size_t)s0 * STRIP_INTS, wl_lds0, tid);

    const char* lut_lds_g = (const char*)smem;   // generic ptr into LDS

    for (int s = s0; s < NUM_STRIPS; s += stride) {
        // consume point: this wave's async copies done, then block barrier
        asm volatile("s_wait_asynccnt 0x0" ::: "memory");
        __syncthreads();

        // prefetch next strip into the other buffer (safe: everyone passed the
        // barrier after finishing compute on that buffer last iteration)
        const int ns = s + stride;
        if (ns < NUM_STRIPS)
            async_stage_strip(walks + (size_t)ns * STRIP_INTS,
                              cur ? wl_lds0 : wl_lds1, tid);

        // strip -> coordinates: 32 tiles = 16 rows x 512 cols of W
        const int tr  = s >> 4;              // tile row   (0..511)
        const int sc  = s & 15;              // strip col  (0..15)
        const int tc8 = sc * 256;            // v2f column base

        const int   i   = tid >> 5;          // row within tile (0..15)
        const int   l32 = tid & 31;
        const int   row = tr * 16 + i;
        const float sl  = W_SCALE_F * sign_l[row];
        const int*  w   = wl_base + (cur ? STRIP_INTS : 0);
        const size_t rb = (size_t)row * 4096 + tc8;

#pragma unroll
        for (int k = 0; k < 8; ++k) {
            const int c2   = l32 + k * 32;                       // 0..255
            // tile t = c2>>3, step = i*8 + (c2&7)
            const int widx = w[((c2 >> 3) << 7) + (i << 3) + (c2 & 7)];
            // branchless per-lane LDS/global routing via FLAT load:
            const v2f* p = (widx < LUT_LDS_N)
                ? (const v2f*)(lut_lds_g + ((uint32_t)widx << 3))
                : (lut2 + widx);
            const v2f lv = *p;                                   // flat_load_b64
            const v2f sr = sign_r2[tc8 + c2];
            v2f o;
            o.x = sl * lv.x * sr.x;
            o.y = sl * lv.y * sr.y;
            // streaming store: output (256MB) must not evict the L2 LUT half
            __builtin_nontemporal_store(o, &out2[rb + c2]);
        }
        cur ^= 1;
    }
}

extern "C" void kernel_launch(void* const* d_in, const int* in_sizes, int n_in,
                              void* d_out, int out_size, void* d_ws, size_t ws_size,
                              hipStream_t stream) {
    (void)in_sizes; (void)n_in; (void)d_ws; (void)ws_size; (void)out_size;

    const int*   walks   = (const int*)d_in[0];
    const v2f*   lut2    = (const v2f*)d_in[1];
    const float* sign_l  = (const float*)d_in[2];
    const v2f*   sign_r2 = (const v2f*)d_in[3];
    v2f*         out2    = (v2f*)d_out;

    dim3 grid(1024);    // 8 strips per block: amortizes the 256KB LUT staging
    dim3 block(512);    // 16 wave32; 288KB LDS -> 1 block per WGP
    trellis_dequant_kernel<<<grid, block, SMEM_BYTES, stream>>>(
        walks, lut2, sign_l, sign_r2, out2);
}